// GELU200_39857296507249
// MI455X (gfx1250) — compile-verified
//
#include <hip/hip_runtime.h>
#include <math.h>

#define TPB 256
#define DSZ 2048   // feature dim; 8 floats per thread

typedef float v4f __attribute__((ext_vector_type(4)));

__device__ __forceinline__ float softplus_f(float v) {
    return fmaxf(v, 0.0f) + log1pf(expf(-fabsf(v)));
}

// CDNA5 hardware tanh (single TRANS op). v_nop covers the documented
// TRANS-op result hazard since the compiler can't see inside the asm.
__device__ __forceinline__ float htanh(float v) {
    float r;
    asm("v_tanh_f32 %0, %1\n\tv_nop" : "=v"(r) : "v"(v));
    return r;
}

// ---------------------------------------------------------------------------
// Prep kernel (single block): per-channel rstd, normalized EMA direction,
// scalar transforms. O(D) work, negligible cost.
// ---------------------------------------------------------------------------
__global__ void __launch_bounds__(TPB)
gelu_gate_prep(const float* __restrict__ mu,
               const float* __restrict__ sq,
               const float* __restrict__ eo,
               const float* __restrict__ ltau,
               const float* __restrict__ lsig,
               const float* __restrict__ lw,
               float* __restrict__ rstd,
               float* __restrict__ eman,
               float* __restrict__ scal,
               int D) {
    const int t = threadIdx.x;
    float s = 0.0f;
    for (int d = t; d < D; d += TPB) {
        float m = mu[d];
        float v = fmaxf(sq[d] - m * m, 1e-4f);
        rstd[d] = 1.0f / (sqrtf(v) + 1e-5f);
        float e = eo[d];
        s += e * e;
    }
    __shared__ float sh[TPB / 32];
    #pragma unroll
    for (int o = 16; o; o >>= 1) s += __shfl_xor(s, o, 32);
    if ((t & 31) == 0) sh[t >> 5] = s;
    __syncthreads();
    if (t == 0) {
        float tot = 0.0f;
        for (int i = 0; i < TPB / 32; ++i) tot += sh[i];
        sh[0] = 1.0f / fmaxf(sqrtf(tot), 1e-12f);
    }
    __syncthreads();
    const float inv = sh[0];
    for (int d = t; d < D; d += TPB) eman[d] = eo[d] * inv;
    if (t == 0) {
        scal[0] = expf(ltau[0]);        // tau
        scal[1] = softplus_f(lsig[0]);  // sigma
        scal[2] = softplus_f(lw[0]);    // w
    }
}

// ---------------------------------------------------------------------------
// Async global -> LDS prefetch of one 8KB row (2 x b128 per thread, GV mode).
// Tracked on ASYNCcnt (2 per call, per wave).
// ---------------------------------------------------------------------------
__device__ __forceinline__ void async_prefetch_row(const float* g0, const float* g1,
                                                   uint32_t l0, uint32_t l1) {
    asm volatile("global_load_async_to_lds_b128 %0, %1, off"
                 :: "v"(l0), "v"(g0) : "memory");
    asm volatile("global_load_async_to_lds_b128 %0, %1, off"
                 :: "v"(l1), "v"(g1) : "memory");
}

// ---------------------------------------------------------------------------
// Main kernel: persistent blocks, one row (D=2048) per iteration, double-
// buffered async LDS staging of x, wave32 shuffle + LDS block reduction.
// ---------------------------------------------------------------------------
__global__ void __launch_bounds__(TPB)
gelu_gate_main(const float* __restrict__ x,
               const float* __restrict__ mu_g,
               const float* __restrict__ rstd_g,
               const float* __restrict__ eman_g,
               const float* __restrict__ scal,
               const float* __restrict__ surp_p,
               float* __restrict__ out,
               int R) {
    __shared__ float lbuf[2][DSZ];          // 16 KB double buffer for x rows
    __shared__ float red[3][TPB / 32];

    const int tid = threadIdx.x;
    const int d0  = tid * 4;                // [0, 1024)
    const int d1  = DSZ / 2 + tid * 4;      // [1024, 2048)

    // Per-thread channel constants, loaded once, reused for every row.
    const float4 mu_a = *(const float4*)(mu_g  + d0);
    const float4 mu_b = *(const float4*)(mu_g  + d1);
    const float4 rs_a = *(const float4*)(rstd_g + d0);
    const float4 rs_b = *(const float4*)(rstd_g + d1);
    const float4 en_a = *(const float4*)(eman_g + d0);
    const float4 en_b = *(const float4*)(eman_g + d1);
    const float tau = scal[0], sigma = scal[1], w = scal[2];
    const float surp_ema = surp_p[0];

    const uint32_t lds_base = (uint32_t)(uintptr_t)(&lbuf[0][0]);
    const uint32_t lt0   = lds_base + (uint32_t)(tid * 16);   // buf0, first half
    const uint32_t lhalf = (uint32_t)(DSZ / 2) * 4u;          // 4096 B
    const uint32_t lrow  = (uint32_t)DSZ * 4u;                // 8192 B

    int row = (int)blockIdx.x;
    const int stride = (int)gridDim.x;
    int buf = 0;

    if (row < R) {
        const float* rp = x + (size_t)row * DSZ;
        async_prefetch_row(rp + d0, rp + d1, lt0, lt0 + lhalf);
    }

    for (; row < R; row += stride) {
        const int nxt = row + stride;
        if (nxt < R) {
            const float* rp = x + (size_t)nxt * DSZ;
            const uint32_t lb = lt0 + (buf ^ 1) * lrow;
            async_prefetch_row(rp + d0, rp + d1, lb, lb + lhalf);
            asm volatile("s_wait_asynccnt 2" ::: "memory");  // current row landed
        } else {
            asm volatile("s_wait_asynccnt 0" ::: "memory");
        }
        __syncthreads();  // barrier A: all waves' async portions visible

        const float4 xa = *(const float4*)(&lbuf[buf][d0]);
        const float4 xb = *(const float4*)(&lbuf[buf][d1]);

        const float xs[8]  = {xa.x, xa.y, xa.z, xa.w, xb.x, xb.y, xb.z, xb.w};
        const float mus[8] = {mu_a.x, mu_a.y, mu_a.z, mu_a.w, mu_b.x, mu_b.y, mu_b.z, mu_b.w};
        const float rss[8] = {rs_a.x, rs_a.y, rs_a.z, rs_a.w, rs_b.x, rs_b.y, rs_b.z, rs_b.w};
        const float ens[8] = {en_a.x, en_a.y, en_a.z, en_a.w, en_b.x, en_b.y, en_b.z, en_b.w};

        float o[8];
        float s_absz = 0.0f, s_oe = 0.0f, s_oo = 0.0f;
        #pragma unroll
        for (int k = 0; k < 8; ++k) {
            const float xv = xs[k];
            const float u  = 0.7978845608028654f * (xv + 0.044715f * xv * xv * xv);
            const float g  = 0.5f * xv * (1.0f + htanh(u));   // hw v_tanh_f32
            o[k] = g;
            const float z = (xv - mus[k]) * rss[k];
            s_absz += fabsf(z);
            s_oe   += g * ens[k];
            s_oo   += g * g;
        }

        // wave32 tree reduction
        #pragma unroll
        for (int m = 16; m; m >>= 1) {
            s_absz += __shfl_xor(s_absz, m, 32);
            s_oe   += __shfl_xor(s_oe,   m, 32);
            s_oo   += __shfl_xor(s_oo,   m, 32);
        }
        if ((tid & 31) == 0) {
            red[0][tid >> 5] = s_absz;
            red[1][tid >> 5] = s_oe;
            red[2][tid >> 5] = s_oo;
        }
        __syncthreads();  // barrier B: partials ready; lbuf[buf] reads done

        // Every thread reduces the 8 wave partials and computes the gate
        // redundantly (cheap with hw trans ops) -> no broadcast barrier.
        float a = 0.0f, b = 0.0f, c = 0.0f;
        #pragma unroll
        for (int i = 0; i < TPB / 32; ++i) {
            a += red[0][i]; b += red[1][i]; c += red[2][i];
        }
        const float mean_absz = a * (1.0f / (float)DSZ);
        const float surp_mod  = fmaxf(htanh(sigma * mean_absz) - surp_ema, 0.0f);
        const float inv_on    = 1.0f / fmaxf(sqrtf(c), 1e-12f);
        const float cs        = fminf(fmaxf(b * inv_on, -1.0f), 1.0f);
        const float gate      = expf(-tau * cs) * (1.0f + w * surp_mod);

        // streaming (non-temporal) output stores
        float* orow = out + (size_t)row * DSZ;
        v4f oa = {o[0] * gate, o[1] * gate, o[2] * gate, o[3] * gate};
        v4f ob = {o[4] * gate, o[5] * gate, o[6] * gate, o[7] * gate};
        __builtin_nontemporal_store(oa, (v4f*)(orow + d0));
        __builtin_nontemporal_store(ob, (v4f*)(orow + d1));

        buf ^= 1;
    }
}

// ---------------------------------------------------------------------------
// Launch
// ---------------------------------------------------------------------------
extern "C" void kernel_launch(void* const* d_in, const int* in_sizes, int n_in,
                              void* d_out, int out_size, void* d_ws, size_t ws_size,
                              hipStream_t stream) {
    const float* x    = (const float*)d_in[0];
    const float* mu   = (const float*)d_in[1];
    const float* sq   = (const float*)d_in[2];
    const float* eo   = (const float*)d_in[3];
    const float* surp = (const float*)d_in[4];
    const float* ltau = (const float*)d_in[5];
    const float* lsig = (const float*)d_in[6];
    const float* lw   = (const float*)d_in[7];

    const int D = in_sizes[1];          // 2048
    const int R = in_sizes[0] / D;      // 16384 rows

    float* ws   = (float*)d_ws;
    float* rstd = ws;                   // [D]
    float* eman = ws + D;               // [D]
    float* scal = ws + 2 * D;           // [3]

    gelu_gate_prep<<<1, TPB, 0, stream>>>(mu, sq, eo, ltau, lsig, lw,
                                          rstd, eman, scal, D);

    int grid = (R < 2048) ? R : 2048;   // persistent blocks, ~8 rows each
    gelu_gate_main<<<grid, TPB, 0, stream>>>(x, mu, rstd, eman, scal, surp,
                                             (float*)d_out, R);
}